// CovCorrLog_22531398434824
// MI455X (gfx1250) — compile-verified
//
#include <hip/hip_runtime.h>
#include <math.h>
#include <stdint.h>

// ---------------------------------------------------------------------------
// MPNCOV pipeline for MI455X (gfx1250), f32 WMMA (V_WMMA_F32_16X16X4_F32).
// Compute-bound: ~1.24 TFLOP of 512^3 f32 matmuls; working set fits in L2.
// eigh-logm replaced by matmul-only inverse scaling-and-squaring + Gregory
// series so every heavy op maps onto the WMMA pipe.  The core matmul uses
// 2x2 register blocking per wave (4 WMMA accumulators) and double-buffered
// async global->LDS staging (GLOBAL_LOAD_ASYNC_TO_LDS_B128 / ASYNCcnt).
// ---------------------------------------------------------------------------

typedef float v2f __attribute__((ext_vector_type(2)));
typedef float v8f __attribute__((ext_vector_type(8)));
typedef int   v4i __attribute__((ext_vector_type(4)));

#define NDIM     512
#define MSZ      262144      /* 512*512 */
#define NBATCH   64
#define MSPAT    784         /* 28*28 */
#define EPSF     1.1920929e-07f
#define EWBLK    65536       /* (64*512*512)/256 elementwise blocks */
#define KC       32
#define NCHUNK   16          /* 512 / KC */

static __device__ __forceinline__ v8f wmma4(v2f a, v2f b, v8f c) {
  // D = A(16x4) * B(4x16) + C(16x16), all f32, wave32
  return __builtin_amdgcn_wmma_f32_16x16x4_f32(false, a, false, b,
                                               (short)0, c, false, false);
}

// --- async global->LDS copy (16B per lane), ASYNCcnt-tracked ---------------
#if __has_builtin(__builtin_amdgcn_global_load_async_to_lds_b128)
#define ASYNC_B128(gp, lp)                                                    \
  __builtin_amdgcn_global_load_async_to_lds_b128((v4i*)(gp), (v4i*)(lp), 0, 0)
#else
#define ASYNC_B128(gp, lp)                                                    \
  asm volatile("global_load_async_to_lds_b128 %0, %1, off"                    \
               :: "v"((uint32_t)(uintptr_t)(lp)),                             \
                  "v"((uint64_t)(uintptr_t)(gp))                              \
               : "memory")
#endif

#if __has_builtin(__builtin_amdgcn_s_wait_asynccnt)
#define WAIT_ASYNC(n) __builtin_amdgcn_s_wait_asynccnt(n)
#else
#define WAIT_ASYNC(n) asm volatile("s_wait_asynccnt %0" :: "i"(n) : "memory")
#endif

// ---------------------------------------------------------------------------
// Per-channel mean over the 784 spatial positions (wave per channel).
// ---------------------------------------------------------------------------
__global__ __launch_bounds__(256) void mean_kernel(const float* __restrict__ x,
                                                   float* __restrict__ mu) {
  int wave = threadIdx.x >> 5, lane = threadIdx.x & 31;
  int ch = blockIdx.x * 8 + wave;              // 0 .. 64*512-1
  const float* p = x + (size_t)ch * MSPAT;
  float s = 0.0f;
  for (int i = lane; i < MSPAT; i += 32) s += p[i];
  for (int off = 16; off; off >>= 1) s += __shfl_down(s, off, 32);
  if (lane == 0) mu[ch] = s * (1.0f / (float)MSPAT);
}

// ---------------------------------------------------------------------------
// Gram: G[b] = X[b] @ X[b]^T, X = (512 x 784). 64x64 tile / block, 8 waves,
// each wave owns two 16x16 WMMA accumulators. K staged in chunks of 28.
// ---------------------------------------------------------------------------
__global__ __launch_bounds__(256) void gram_kernel(const float* __restrict__ x,
                                                   float* __restrict__ G) {
  __shared__ float As[64][36];
  __shared__ float Bs[64][36];
  int b = blockIdx.z;
  int rowBase = blockIdx.x * 64, colBase = blockIdx.y * 64;
  const float* xb = x + (size_t)b * NDIM * MSPAT;
  int t = threadIdx.x, wave = t >> 5, lane = t & 31;
  int si = (wave & 3) * 16, sj = (wave >> 2) * 32;
  int hk = (lane >> 4) << 1, lr = lane & 15;
  v8f acc0 = {}, acc1 = {};
  for (int k0 = 0; k0 < MSPAT; k0 += 28) {
#pragma unroll
    for (int i = 0; i < 2; ++i) {
      int idx = t + i * 256;                    // 448 float4 per tile
      if (idx < 448) {
        int r = idx / 7, kk = (idx % 7) << 2;
        *(float4*)&As[r][kk] =
            *(const float4*)(xb + (size_t)(rowBase + r) * MSPAT + k0 + kk);
        *(float4*)&Bs[r][kk] =
            *(const float4*)(xb + (size_t)(colBase + r) * MSPAT + k0 + kk);
      }
    }
    __syncthreads();
#pragma unroll
    for (int kk = 0; kk < 28; kk += 4) {
      v2f a  = *(const v2f*)&As[si + lr][kk + hk];
      v2f b0 = *(const v2f*)&Bs[sj + lr][kk + hk];
      v2f b1 = *(const v2f*)&Bs[sj + 16 + lr][kk + hk];
      acc0 = wmma4(a, b0, acc0);
      acc1 = wmma4(a, b1, acc1);
    }
    __syncthreads();
  }
  float* Gb = G + (size_t)b * MSZ;
  int gr = rowBase + si + ((lane >> 4) << 3);
  int gc0 = colBase + sj + lr, gc1 = gc0 + 16;
#pragma unroll
  for (int v = 0; v < 8; ++v) {
    Gb[(size_t)(gr + v) * NDIM + gc0] = acc0[v];
    Gb[(size_t)(gr + v) * NDIM + gc1] = acc1[v];
  }
}

// ---------------------------------------------------------------------------
// Generic batched 512^3 WMMA matmul: D = alpha * s * (A@B) + beta * I.
// 128x64 block tile, 8 waves, each wave computes a 32x32 block = 4 WMMA
// accumulators (A/B fragments each reused twice).  K in chunks of 32,
// double-buffered LDS filled by async global->LDS B128 copies; chunk c+2 is
// issued after chunk c is consumed; s_wait_asynccnt(6)+barrier synchronizes.
// ---------------------------------------------------------------------------
__global__ __launch_bounds__(256) void mm512_kernel(
    const float* __restrict__ A, const float* __restrict__ Bm,
    float* __restrict__ Dm, float alpha, float beta,
    const float* __restrict__ scal, int useScale) {
  __shared__ float As[2][128][36];   // [buf][row][k]
  __shared__ float Bs[2][32][68];    // [buf][k][n]
  int b = blockIdx.z;
  const float* Ab = A + (size_t)b * MSZ;
  const float* Bb = Bm + (size_t)b * MSZ;
  float* Db = Dm + (size_t)b * MSZ;
  int t = threadIdx.x, wave = t >> 5, lane = t & 31;
  int rowBase = blockIdx.x * 128, colBase = blockIdx.y * 64;
  int wr = (wave & 3) * 32;          // wave's 32-row strip
  int wc = (wave >> 2) * 32;         // wave's 32-col strip
  int hk = (lane >> 4) << 1, lr = lane & 15;

  v8f acc00 = {}, acc01 = {}, acc10 = {}, acc11 = {};

  auto issue = [&](int c, int buf) {
    int k0 = c * KC;
#pragma unroll
    for (int i = 0; i < 4; ++i) {                 // A: 128x32 = 1024 float4
      int idx = t + i * 256;
      int r = idx >> 3, kk = (idx & 7) << 2;
      ASYNC_B128(Ab + (size_t)(rowBase + r) * NDIM + k0 + kk, &As[buf][r][kk]);
    }
#pragma unroll
    for (int i = 0; i < 2; ++i) {                 // B: 32x64 = 512 float4
      int idx = t + i * 256;
      int r = idx >> 4, nn = (idx & 15) << 2;
      ASYNC_B128(Bb + (size_t)(k0 + r) * NDIM + colBase + nn, &Bs[buf][r][nn]);
    }
  };

  auto compute = [&](int buf) {
#pragma unroll
    for (int kk = 0; kk < KC; kk += 4) {
      v2f a0 = *(const v2f*)&As[buf][wr + lr][kk + hk];
      v2f a1 = *(const v2f*)&As[buf][wr + 16 + lr][kk + hk];
      v2f b0, b1;
      b0.x = Bs[buf][kk + hk][wc + lr];      b0.y = Bs[buf][kk + hk + 1][wc + lr];
      b1.x = Bs[buf][kk + hk][wc + 16 + lr]; b1.y = Bs[buf][kk + hk + 1][wc + 16 + lr];
      acc00 = wmma4(a0, b0, acc00);
      acc01 = wmma4(a0, b1, acc01);
      acc10 = wmma4(a1, b0, acc10);
      acc11 = wmma4(a1, b1, acc11);
    }
  };

  issue(0, 0);
  issue(1, 1);
#pragma unroll 1
  for (int c = 0; c < NCHUNK - 1; ++c) {
    WAIT_ASYNC(6);                    // chunk c's 6 copies done (c+1 in flight)
    __syncthreads();
    compute(c & 1);
    __syncthreads();                  // everyone done reading buf (c&1)
    if (c + 2 < NCHUNK) issue(c + 2, c & 1);
  }
  WAIT_ASYNC(0);
  __syncthreads();
  compute((NCHUNK - 1) & 1);

  float s = alpha * (useScale ? scal[b] : 1.0f);
  int gr0 = rowBase + wr + ((lane >> 4) << 3);
  int gr1 = gr0 + 16;
  int gc0 = colBase + wc + lr, gc1 = gc0 + 16;
#pragma unroll
  for (int v = 0; v < 8; ++v) {
    int gi0 = gr0 + v, gi1 = gr1 + v;
    float d00 = s * acc00[v]; if (gi0 == gc0) d00 += beta;
    float d01 = s * acc01[v]; if (gi0 == gc1) d01 += beta;
    float d10 = s * acc10[v]; if (gi1 == gc0) d10 += beta;
    float d11 = s * acc11[v]; if (gi1 == gc1) d11 += beta;
    Db[(size_t)gi0 * NDIM + gc0] = d00;
    Db[(size_t)gi0 * NDIM + gc1] = d01;
    Db[(size_t)gi1 * NDIM + gc0] = d10;
    Db[(size_t)gi1 * NDIM + gc1] = d11;
  }
}

// ---------------------------------------------------------------------------
// Elementwise / reduction helpers.
// ---------------------------------------------------------------------------
__global__ void cov_finalize_kernel(float* __restrict__ G,
                                    const float* __restrict__ mu) {
  size_t idx = (size_t)blockIdx.x * 256 + threadIdx.x;
  int b = (int)(idx >> 18);
  int ij = (int)(idx & (MSZ - 1));
  int i = ij >> 9, j = ij & 511;
  G[idx] = G[idx] * (1.0f / (float)MSPAT) - mu[b * NDIM + i] * mu[b * NDIM + j];
}

__global__ void trace_kernel(const float* __restrict__ A,
                             float* __restrict__ tr, float* __restrict__ sc) {
  __shared__ float red[256];
  int b = blockIdx.x, t = threadIdx.x;
  const float* Ab = A + (size_t)b * MSZ;
  red[t] = Ab[(size_t)t * 513] + Ab[(size_t)(t + 256) * 513];
  __syncthreads();
  for (int off = 128; off; off >>= 1) {
    if (t < off) red[t] += red[t + off];
    __syncthreads();
  }
  if (t == 0) { tr[b] = red[0]; sc[b] = sqrtf(red[0]); }
}

__global__ void invtrace_scale_kernel(const float* __restrict__ A,
                                      float* __restrict__ D,
                                      const float* __restrict__ tr) {
  size_t idx = (size_t)blockIdx.x * 256 + threadIdx.x;
  int b = (int)(idx >> 18);
  D[idx] = A[idx] / tr[b];
}

__global__ void axpbyI_kernel(const float* __restrict__ A, float* __restrict__ D,
                              float a, float bI) {
  size_t idx = (size_t)blockIdx.x * 256 + threadIdx.x;
  int ij = (int)(idx & (MSZ - 1));
  int i = ij >> 9, j = ij & 511;
  float v = a * A[idx];
  if (i == j) v += bI;
  D[idx] = v;
}

__global__ void diag_rstd_kernel(const float* __restrict__ S,
                                 float* __restrict__ rstd) {
  int idx = blockIdx.x * 256 + threadIdx.x;   // 64*512
  int b = idx >> 9, i = idx & 511;
  float d = S[(size_t)b * MSZ + (size_t)i * 513];
  d = d < EPSF ? EPSF : d;
  rstd[idx] = rsqrtf(d);
}

__global__ void corr_kernel(float* __restrict__ S,
                            const float* __restrict__ rstd) {
  size_t idx = (size_t)blockIdx.x * 256 + threadIdx.x;
  int b = (int)(idx >> 18);
  int ij = (int)(idx & (MSZ - 1));
  int i = ij >> 9, j = ij & 511;
  S[idx] = S[idx] * rstd[b * NDIM + i] * rstd[b * NDIM + j];
}

__global__ void combine3_kernel(const float* __restrict__ Z1,
                                const float* __restrict__ Z3,
                                const float* __restrict__ Z5,
                                float* __restrict__ L, float c) {
  size_t idx = (size_t)blockIdx.x * 256 + threadIdx.x;
  L[idx] = c * (Z1[idx] + Z3[idx] * (1.0f / 3.0f) + Z5[idx] * (1.0f / 5.0f));
}

__global__ void triu_kernel(const float* __restrict__ L, float* __restrict__ out) {
  int j = blockIdx.x * 256 + threadIdx.x;
  int i = blockIdx.y, b = blockIdx.z;
  if (j < i || j >= NDIM) return;
  size_t rowOff = (size_t)i * NDIM - ((size_t)i * (i - 1)) / 2;
  float v = (i == j) ? 0.0f : 2.0f * L[(size_t)b * MSZ + (size_t)i * NDIM + j];
  out[(size_t)b * 131328 + rowOff + (size_t)(j - i)] = v;
}

// ---------------------------------------------------------------------------
// Host-side orchestration.
// ---------------------------------------------------------------------------
static inline void launch_mm(hipStream_t s, const float* A, const float* B,
                             float* D, float alpha, float beta,
                             const float* scal, int useScale) {
  mm512_kernel<<<dim3(4, 8, NBATCH), 256, 0, s>>>(A, B, D, alpha, beta, scal,
                                                  useScale);
}

// MPNCOV Newton-Schulz sqrt: Sout = sqrtm(Ain). t5 may alias Ain.
static void run_sqrt_ns(hipStream_t s, const float* Ain, float* Sout,
                        float* t1, float* t2, float* t3, float* t4, float* t5,
                        float* tr, float* sc) {
  trace_kernel<<<NBATCH, 256, 0, s>>>(Ain, tr, sc);
  invtrace_scale_kernel<<<EWBLK, 256, 0, s>>>(Ain, t1, tr);       // An
  axpbyI_kernel<<<EWBLK, 256, 0, s>>>(t1, t2, -0.5f, 1.5f);       // Z0
  launch_mm(s, t1, t2, t3, 1.0f, 0.0f, nullptr, 0);               // Y0 = An@Z0
  float* Y = t3; float* Z = t2; float* Ys = t1; float* Zs = t5;
  for (int it = 0; it < 3; ++it) {
    launch_mm(s, Z, Y, t4, -0.5f, 1.5f, nullptr, 0);              // T = 1.5I-.5ZY
    launch_mm(s, Y, t4, Ys, 1.0f, 0.0f, nullptr, 0);              // Ynew
    launch_mm(s, t4, Z, Zs, 1.0f, 0.0f, nullptr, 0);              // Znew
    float* tmp = Y; Y = Ys; Ys = tmp;
    tmp = Z; Z = Zs; Zs = tmp;
  }
  launch_mm(s, Z, Y, t4, -0.5f, 1.5f, nullptr, 0);
  launch_mm(s, Y, t4, Sout, 1.0f, 0.0f, sc, 1);                   // * sqrt(trace)
}

extern "C" void kernel_launch(void* const* d_in, const int* in_sizes, int n_in,
                              void* d_out, int out_size, void* d_ws,
                              size_t ws_size, hipStream_t stream) {
  (void)in_sizes; (void)n_in; (void)out_size; (void)ws_size;
  const float* x = (const float*)d_in[0];
  float* out = (float*)d_out;
  float* base = (float*)d_ws;
  const size_t BANKF = (size_t)NBATCH * MSZ;   // floats per matrix bank (64 MB)
  float* W0 = base + 0 * BANKF;
  float* W1 = base + 1 * BANKF;
  float* W2 = base + 2 * BANKF;
  float* W3 = base + 3 * BANKF;
  float* W4 = base + 4 * BANKF;
  float* W5 = base + 5 * BANKF;
  float* mu = base + 6 * BANKF;                // 64*512
  float* tr = mu + NBATCH * NDIM;              // 64
  float* sc = tr + NBATCH;                     // 64
  float* rstd = sc + NBATCH;                   // 64*512

  // 1) Covariance pooling: C = X@X^T / M - mu mu^T
  mean_kernel<<<4096, 256, 0, stream>>>(x, mu);
  gram_kernel<<<dim3(8, 8, NBATCH), 256, 0, stream>>>(x, W0);
  cov_finalize_kernel<<<EWBLK, 256, 0, stream>>>(W0, mu);

  // 2) Matrix square root (Newton-Schulz, 5 iters) -> W5
  run_sqrt_ns(stream, W0, W5, W1, W2, W3, W4, W0, tr, sc);

  // 3) Correlation normalization (in place on W5)
  diag_rstd_kernel<<<128, 256, 0, stream>>>(W5, rstd);
  corr_kernel<<<EWBLK, 256, 0, stream>>>(W5, rstd);

  // 4) logm via inverse scaling-and-squaring: 4 nested NS sqrts
  run_sqrt_ns(stream, W5, W4, W0, W1, W2, W3, W5, tr, sc);
  run_sqrt_ns(stream, W4, W5, W0, W1, W2, W3, W4, tr, sc);
  run_sqrt_ns(stream, W5, W4, W0, W1, W2, W3, W5, tr, sc);
  run_sqrt_ns(stream, W4, W5, W0, W1, W2, W3, W4, tr, sc);
  // A = R^(1/16) in W5

  // 5) Gregory series: log(A) = 2(Zg + Zg^3/3 + Zg^5/5), Zg = (A-I)(A+I)^-1
  axpbyI_kernel<<<EWBLK, 256, 0, stream>>>(W5, W0, 1.0f, 1.0f);   // P = A + I
  axpbyI_kernel<<<EWBLK, 256, 0, stream>>>(W5, W1, 1.0f, -1.0f);  // N = A - I
  axpbyI_kernel<<<EWBLK, 256, 0, stream>>>(W5, W2, 0.0f, 0.5f);   // X0 = 0.5 I
  float* X = W2; float* Xs = W4;
  for (int it = 0; it < 4; ++it) {            // NS inverse of P (P ~ 2I)
    launch_mm(stream, W0, X, W3, -1.0f, 2.0f, nullptr, 0);        // T = 2I - PX
    launch_mm(stream, X, W3, Xs, 1.0f, 0.0f, nullptr, 0);         // Xnew = X T
    float* tmp = X; X = Xs; Xs = tmp;
  }                                            // X = P^-1 (in W2)
  launch_mm(stream, W1, X, W5, 1.0f, 0.0f, nullptr, 0);           // Zg  -> W5
  launch_mm(stream, W5, W5, W0, 1.0f, 0.0f, nullptr, 0);          // Zg^2-> W0
  launch_mm(stream, W0, W5, W1, 1.0f, 0.0f, nullptr, 0);          // Zg^3-> W1
  launch_mm(stream, W1, W0, W3, 1.0f, 0.0f, nullptr, 0);          // Zg^5-> W3
  // logC = 2^4 * 2 * (Zg + Zg^3/3 + Zg^5/5)  -> W2
  combine3_kernel<<<EWBLK, 256, 0, stream>>>(W5, W1, W3, W2, 32.0f);

  // 6) olm (zero diag, off + off^T = 2*off) + row-major triuvec
  triu_kernel<<<dim3(2, NDIM, NBATCH), 256, 0, stream>>>(W2, out);
}